// MultivariateMuyGPs_layer_59390807769351
// MI455X (gfx1250) — compile-verified
//
#include <hip/hip_runtime.h>
#include <hip/hip_bf16.h>
#include <math.h>

// MuyGPs multivariate layer for MI455X (gfx1250, wave32).
// One workgroup (256 threads = 8 waves) per batch element.
// Gram matrix via V_WMMA_F32_16X16X4_F32; Cholesky + 2-RHS solves in LDS.

typedef __attribute__((ext_vector_type(2))) float v2f;
typedef __attribute__((ext_vector_type(8))) float v8f;

#define FEAT   40
#define NN     64
#define MDL    10
#define BATCHN 2048
#define XS_ST  41   // padded LDS row stride for xn (floats) -> odd, conflict-free
#define K_ST   65   // padded LDS row stride for 64x64 matrices

__global__ void __launch_bounds__(256)
muygps_zero_sigma(float* sigma) {
    if (threadIdx.x < MDL) sigma[threadIdx.x] = 0.0f;
}

__global__ void __launch_bounds__(256)
muygps_kernel(const float* __restrict__ x,
              const float* __restrict__ length_scale,
              const float* __restrict__ eps,
              const int*   __restrict__ batch_indices,
              const int*   __restrict__ nn_indices,
              const float* __restrict__ nn_targets,
              float* __restrict__ out_pred,
              float* __restrict__ out_var,
              float* __restrict__ out_sigma)
{
    __shared__ float xs[NN * XS_ST];   // gathered neighbor features (64 x 40)
    __shared__ float xb[FEAT];         // center point features
    __shared__ float n2[NN];           // squared norms of neighbors
    __shared__ float pd[NN * K_ST];    // pairwise l2 distances (64 x 64)
    __shared__ float cd[NN];           // crosswise l2 distances
    __shared__ float Kp[NN * K_ST];    // kernel matrix -> in-place Cholesky L
    __shared__ float kc[NN];           // cross kernel (per model)
    __shared__ float ybuf[NN];         // targets (per model)
    __shared__ float fr[2][NN];        // RHS / solution buffers (F and G)
    __shared__ float red[3 * NN];      // reduction scratch

    const int tid = threadIdx.x;
    const int b   = blockIdx.x;

    // ---- gather xn rows (via nn indices) and xb into LDS ----
    {
        const int row = tid >> 2;          // 0..63
        const int sub = tid & 3;
        const int src = nn_indices[b * NN + row];
        for (int f = sub; f < FEAT; f += 4)
            xs[row * XS_ST + f] = x[src * FEAT + f];
        if (tid < FEAT)
            xb[tid] = x[batch_indices[b] * FEAT + tid];
    }
    __syncthreads();

    // ---- squared norms + crosswise distances (64 lanes, 40-deep dot) ----
    if (tid < NN) {
        float s = 0.0f, c = 0.0f;
        for (int f = 0; f < FEAT; ++f) {
            const float v = xs[tid * XS_ST + f];
            s += v * v;
            const float d = xb[f] - v;
            c += d * d;
        }
        n2[tid] = s;
        cd[tid] = sqrtf(fmaxf(c, 0.0f));
    }
    __syncthreads();

    // ---- pairwise Gram Xn*Xn^T via f32 WMMA, then distances ----
    // 64x64 output = 16 tiles of 16x16; 8 waves x 2 tiles each; K=40 = 10 steps of 4.
    {
        const int wave = tid >> 5;
        const int lane = tid & 31;
        const int lo   = lane & 15;   // row-in-tile for A/B fragments, col for C/D
        const int hi   = lane >> 4;   // K-half select for A/B, M+8 select for C/D
        for (int t = wave * 2; t < wave * 2 + 2; ++t) {
            const int i0 = (t >> 2) * 16;
            const int j0 = (t & 3) * 16;
            v8f acc = {0.f, 0.f, 0.f, 0.f, 0.f, 0.f, 0.f, 0.f};
            const float* ra = &xs[(i0 + lo) * XS_ST];
            const float* rb = &xs[(j0 + lo) * XS_ST];
#pragma unroll
            for (int k0 = 0; k0 < FEAT; k0 += 4) {
                const int ka = k0 + 2 * hi;
                v2f a  = { ra[ka], ra[ka + 1] };   // A: 16x4, row=lo, K=2*hi,2*hi+1
                v2f bb = { rb[ka], rb[ka + 1] };   // B: 4x16 = Xn^T slice (symmetric load)
                acc = __builtin_amdgcn_wmma_f32_16x16x4_f32(
                        false, a, false, bb, (short)0, acc, false, false);
            }
            // C/D layout: VGPR r, lanes 0-15 -> M=r, lanes 16-31 -> M=r+8
#pragma unroll
            for (int r = 0; r < 8; ++r) {
                const int gi = i0 + r + 8 * hi;
                const int gj = j0 + lo;
                const float d2 = n2[gi] + n2[gj] - 2.0f * acc[r];
                pd[gi * K_ST + gj] = sqrtf(fmaxf(d2, 0.0f));
            }
        }
    }
    __syncthreads();

    const float inv_bn = 1.0f / (float)(BATCHN * NN);

    for (int m = 0; m < MDL; ++m) {
        const float inv_ls = 1.0f / length_scale[m];
        const float epsm   = eps[m];

        // ---- build Kp = exp(-pd/ls) + eps*I ----
        for (int e = tid; e < NN * NN; e += 256) {
            const int i = e >> 6, j = e & 63;
            float v = __expf(-pd[i * K_ST + j] * inv_ls);
            if (i == j) v += epsm;
            Kp[i * K_ST + j] = v;
        }
        if (tid < NN) {
            const float kcv = __expf(-cd[tid] * inv_ls);
            const float yv  = nn_targets[(b * NN + tid) * MDL + m];
            kc[tid]    = kcv;
            ybuf[tid]  = yv;
            fr[0][tid] = yv;    // RHS 0: targets  -> F
            fr[1][tid] = kcv;   // RHS 1: Kcross   -> G
        }
        __syncthreads();

        // ---- in-place Cholesky (lower), 64 elimination steps ----
        for (int k = 0; k < NN; ++k) {
            if (tid == 0)
                Kp[k * K_ST + k] = sqrtf(Kp[k * K_ST + k]);
            __syncthreads();
            if (tid < NN && tid > k)
                Kp[tid * K_ST + k] /= Kp[k * K_ST + k];
            __syncthreads();
            {   // rank-1 trailing update: 64 rows x 4 column-threads
                const int i  = tid >> 2;
                const int js = tid & 3;
                if (i > k) {
                    const float lik = Kp[i * K_ST + k];
                    for (int j = k + 1 + js; j <= i; j += 4)
                        Kp[i * K_ST + j] -= lik * Kp[j * K_ST + k];
                }
            }
            __syncthreads();
        }

        // ---- solve Kp f = rhs for both RHS simultaneously ----
        const int rr = tid >> 6;   // 0,1 active; 2,3 idle (barriers only)
        const int ii = tid & 63;
        // forward: L z = rhs
        for (int k = 0; k < NN; ++k) {
            if (rr < 2 && ii == k) fr[rr][k] /= Kp[k * K_ST + k];
            __syncthreads();
            if (rr < 2 && ii > k)  fr[rr][ii] -= Kp[ii * K_ST + k] * fr[rr][k];
            __syncthreads();
        }
        // backward: L^T f = z   (L^T[i][k] = L[k][i])
        for (int k = NN - 1; k >= 0; --k) {
            if (rr < 2 && ii == k) fr[rr][k] /= Kp[k * K_ST + k];
            __syncthreads();
            if (rr < 2 && ii < k)  fr[rr][ii] -= Kp[k * K_ST + ii] * fr[rr][k];
            __syncthreads();
        }

        // ---- reductions: predictions, variances, sigma_sq ----
        if (tid < NN) {
            red[tid]          = kc[tid]   * fr[0][tid];   // Kcross . F
            red[NN + tid]     = kc[tid]   * fr[1][tid];   // Kcross . G
            red[2 * NN + tid] = ybuf[tid] * fr[0][tid];   // Y . F
        }
        __syncthreads();
        if (tid == 0) {
            float s0 = 0.f, s1 = 0.f, s2 = 0.f;
            for (int j = 0; j < NN; ++j) {
                s0 += red[j];
                s1 += red[NN + j];
                s2 += red[2 * NN + j];
            }
            out_pred[b * MDL + m] = s0;
            out_var[b * MDL + m]  = 1.0f - s1;
            atomicAdd(&out_sigma[m], s2 * inv_bn);
        }
        __syncthreads();
    }
}

extern "C" void kernel_launch(void* const* d_in, const int* in_sizes, int n_in,
                              void* d_out, int out_size, void* d_ws, size_t ws_size,
                              hipStream_t stream) {
    const float* x   = (const float*)d_in[0];
    const float* ls  = (const float*)d_in[1];
    const float* ep  = (const float*)d_in[2];
    const int*   bi  = (const int*)  d_in[3];
    const int*   bni = (const int*)  d_in[4];
    const float* tgt = (const float*)d_in[5];

    float* out   = (float*)d_out;
    float* pred  = out;                       // (2048, 10)
    float* var   = out + BATCHN * MDL;        // (2048, 10)
    float* sigma = out + 2 * BATCHN * MDL;    // (10,)

    muygps_zero_sigma<<<1, 256, 0, stream>>>(sigma);
    muygps_kernel<<<BATCHN, 256, 0, stream>>>(x, ls, ep, bi, bni, tgt,
                                              pred, var, sigma);
}